// Model_15805479649900
// MI455X (gfx1250) — compile-verified
//
#include <hip/hip_runtime.h>
#include <math.h>

// ---------------- problem constants ----------------
#define NNODES   124
#define NB       256
#define ND       32
#define NTOT     31744      // NB*NNODES
#define EDG      507904     // NB*NNODES*16
#define ETOT     539648     // EDG + NTOT (self loops)
#define NH       4
#define NC       32
#define HC       128        // NH*NC
#define NL       3
#define K1       3968       // NNODES*NC
#define N1       124
#define TMPF     316        // 124 + 64*3
#define NCLS     40
#define KSPLIT   16
#define KCH      (K1 / KSPLIT)   // 248

typedef __attribute__((ext_vector_type(2))) float v2f;
typedef __attribute__((ext_vector_type(8))) float v8f;

// ---------------- helpers ----------------
__device__ __forceinline__ void atomicMaxF(float* addr, float val) {
  if (!(val < 0.0f))
    atomicMax((int*)addr, __float_as_int(val));
  else
    atomicMin((unsigned int*)addr, (unsigned int)__float_as_int(val));
}

__global__ void fill_kernel(float* __restrict__ p, int n, float v) {
  int t = blockIdx.x * blockDim.x + threadIdx.x;
  if (t < n) p[t] = v;
}

// ---------------- GEMM: H[N,128] = X[N,32] @ W[32,128]  (WMMA f32 16x16x4) --------
__global__ void gemm_xw_kernel(const float* __restrict__ X, const float* __restrict__ W,
                               float* __restrict__ H, int Mtiles) {
  int wid  = (blockIdx.x * blockDim.x + threadIdx.x) >> 5;
  int lane = threadIdx.x & 31;
  if (wid >= Mtiles * 8) return;             // wave-uniform
  int tn = wid & 7;                          // 8 column tiles of 16 (HC=128)
  int tm = wid >> 3;
  int hi  = lane >> 4;                       // lane half selects K pair
  int lid = lane & 15;
  const float* Arow = X + (tm * 16 + lid) * ND;     // A: M = lane&15
  const float* Bcol = W + tn * 16 + lid;            // B: Ncol = lane&15
  v8f acc = {};
#pragma unroll
  for (int kk = 0; kk < 8; ++kk) {
    int kb = kk * 4 + 2 * hi;
    v2f a; a.x = Arow[kb];        a.y = Arow[kb + 1];
    v2f b; b.x = Bcol[kb * HC];   b.y = Bcol[(kb + 1) * HC];
    acc = __builtin_amdgcn_wmma_f32_16x16x4_f32(false, a, false, b,
                                                (short)0, acc, false, false);
  }
  float* O = H + (tm * 16) * HC + tn * 16 + lid;
#pragma unroll
  for (int r = 0; r < 8; ++r)
    O[(r + 8 * hi) * HC] = acc[r];
}

// ------- GEMM (K-split): Yp[kc][256,128] = X[256,kchunk] @ W[kchunk,124] (WMMA) ----
__global__ void gemm_lin1_kernel(const float* __restrict__ X, const float* __restrict__ W,
                                 float* __restrict__ Yp) {
  int wid  = (blockIdx.x * blockDim.x + threadIdx.x) >> 5;
  int lane = threadIdx.x & 31;
  if (wid >= 16 * 8 * KSPLIT) return;        // 16 M-tiles x 8 N-tiles x KSPLIT
  int kc   = wid & (KSPLIT - 1);
  int tile = wid >> 4;
  int tn = tile & 7, tm = tile >> 3;
  int hi = lane >> 4, lid = lane & 15;
  int gcol = tn * 16 + lid;
  bool colok = gcol < N1;
  const float* Arow = X + (tm * 16 + lid) * K1;
  int kstart = kc * KCH;
  v8f acc = {};
  for (int k0 = kstart; k0 < kstart + KCH; k0 += 4) {
    int kb = k0 + 2 * hi;
    __builtin_prefetch(&Arow[kb + 32], 0, 1);
    v2f a; a.x = Arow[kb]; a.y = Arow[kb + 1];
    v2f b;
    if (colok) {
      __builtin_prefetch(&W[(kb + 8) * N1 + gcol], 0, 1);
      b.x = W[kb * N1 + gcol]; b.y = W[(kb + 1) * N1 + gcol];
    } else {
      b.x = 0.0f; b.y = 0.0f;
    }
    acc = __builtin_amdgcn_wmma_f32_16x16x4_f32(false, a, false, b,
                                                (short)0, acc, false, false);
  }
  // partial tile store, padded to 128 columns (cols >=124 are zero partials)
  float* O = Yp + kc * NB * 128 + (tm * 16) * 128 + tn * 16 + lid;
#pragma unroll
  for (int r = 0; r < 8; ++r)
    O[(r + 8 * hi) * 128] = acc[r];
}

__global__ void lin1_reduce_kernel(const float* __restrict__ Yp,
                                   const float* __restrict__ bias,
                                   float* __restrict__ Y) {
  int t = blockIdx.x * blockDim.x + threadIdx.x;
  if (t >= NB * N1) return;
  int r = t / N1, c = t - r * N1;
  float s = bias[c];
#pragma unroll
  for (int k = 0; k < KSPLIT; ++k) s += Yp[k * NB * 128 + r * 128 + c];
  Y[t] = s;
}

// ---------------- edge attention: raw alpha + segment max -------------------------
__global__ void edge_alpha_kernel(const float* __restrict__ H,
                                  const int* __restrict__ srcA, const int* __restrict__ dstA,
                                  const float* __restrict__ attl, const float* __restrict__ attr,
                                  float* __restrict__ alpha, float* __restrict__ amax) {
  int wid  = (blockIdx.x * blockDim.x + threadIdx.x) >> 5;
  int lane = threadIdx.x & 31;
  if (wid >= ETOT) return;
  int s, d;
  if (wid < EDG) { s = srcA[wid]; d = dstA[wid]; } else { s = d = wid - EDG; }
  const float* hs = H + s * HC;
  const float* hd = H + d * HC;
#pragma unroll
  for (int h = 0; h < NH; ++h) {
    float xj  = hs[h * NC + lane];
    float xi  = hd[h * NC + lane];
    float dot = xi * xj;
    float aL  = xj * attl[h * NC + lane];
    float aR  = xi * attr[h * NC + lane];
#pragma unroll
    for (int o = 16; o > 0; o >>= 1) {
      dot += __shfl_xor(dot, o, 32);
      aL  += __shfl_xor(aL, o, 32);
      aR  += __shfl_xor(aR, o, 32);
    }
    if (lane == 0) {
      float sig = 1.0f / (1.0f + expf(-dot));
      float a = (aL + aR) * sig;
      a = (a > 0.0f) ? a : 0.2f * a;         // leaky relu
      alpha[wid * NH + h] = a;
      atomicMaxF(&amax[d * NH + h], a);
    }
  }
}

// ---------------- exp + segment sum ----------------------------------------------
__global__ void edge_exp_kernel(const int* __restrict__ dstA, float* __restrict__ alpha,
                                const float* __restrict__ amax, float* __restrict__ denom) {
  int t = blockIdx.x * blockDim.x + threadIdx.x;
  if (t >= ETOT * NH) return;
  int e = t >> 2, h = t & 3;
  int d = (e < EDG) ? dstA[e] : (e - EDG);
  float ex = expf(alpha[t] - amax[d * NH + h]);
  alpha[t] = ex;
  atomicAdd(&denom[d * NH + h], ex);
}

// ---------------- weighted aggregation -------------------------------------------
__global__ void edge_agg_kernel(const float* __restrict__ H,
                                const int* __restrict__ srcA, const int* __restrict__ dstA,
                                const float* __restrict__ ex, const float* __restrict__ denom,
                                float* __restrict__ agg) {
  int wid  = (blockIdx.x * blockDim.x + threadIdx.x) >> 5;
  int lane = threadIdx.x & 31;
  if (wid >= ETOT) return;
  int s, d;
  if (wid < EDG) { s = srcA[wid]; d = dstA[wid]; } else { s = d = wid - EDG; }
  const float* hs = H + s * HC;
#pragma unroll
  for (int h = 0; h < NH; ++h) {
    float a = ex[wid * NH + h] / (denom[d * NH + h] + 1e-16f);
    atomicAdd(&agg[d * HC + h * NC + lane], hs[h * NC + lane] * a);
  }
}

// ---------------- head mean + bias + BN partial statistics ------------------------
__global__ void node_post_kernel(const float* __restrict__ agg, const float* __restrict__ bias,
                                 float* __restrict__ xout, float* __restrict__ part) {
  __shared__ float ssum[256], ssq[256];
  int lane = threadIdx.x & 31, w = threadIdx.x >> 5;
  float s = 0.0f, q = 0.0f;
  for (int n = blockIdx.x * 8 + w; n < NTOT; n += gridDim.x * 8) {
    const float* p = agg + n * HC;
    float v = 0.25f * (p[lane] + p[NC + lane] + p[2 * NC + lane] + p[3 * NC + lane])
              + bias[lane];
    xout[n * NC + lane] = v;
    s += v; q += v * v;
  }
  ssum[threadIdx.x] = s; ssq[threadIdx.x] = q;
  __syncthreads();
  if (w == 0) {
    float ts = 0.0f, tq = 0.0f;
#pragma unroll
    for (int i = 0; i < 8; ++i) { ts += ssum[i * 32 + lane]; tq += ssq[i * 32 + lane]; }
    part[blockIdx.x * 64 + lane]      = ts;   // deterministic per-block partials
    part[blockIdx.x * 64 + 32 + lane] = tq;
  }
}

__global__ void bn_finalize_kernel(const float* __restrict__ part,
                                   const float* __restrict__ gamma,
                                   const float* __restrict__ beta,
                                   float* __restrict__ stats) {
  int c = threadIdx.x;
  if (c >= NC) return;
  float s = 0.0f, q = 0.0f;
  for (int i = 0; i < 64; ++i) { s += part[i * 64 + c]; q += part[i * 64 + 32 + c]; }
  float mu  = s / (float)NTOT;
  float var = q / (float)NTOT - mu * mu;
  float sc  = rsqrtf(var + 1e-5f) * gamma[c];
  stats[c]      = sc;
  stats[NC + c] = beta[c] - mu * sc;
}

// -------- BN apply + ReLU + graph mean/max pooling (wave per graph, no atomics) ---
__global__ void bn_relu_pool_kernel(float* __restrict__ xbuf, const float* __restrict__ stats,
                                    float* __restrict__ inter_l) {
  int wid  = (blockIdx.x * blockDim.x + threadIdx.x) >> 5;
  int lane = threadIdx.x & 31;
  if (wid >= NB) return;
  float sc = stats[lane], sh = stats[NC + lane];
  float sum = 0.0f, mx = 0.0f;
  float* p = xbuf + wid * NNODES * NC;
  for (int n = 0; n < NNODES; ++n) {
    float v = p[n * NC + lane] * sc + sh;
    v = (v > 0.0f) ? v : 0.0f;
    p[n * NC + lane] = v;
    sum += v;
    mx = (v > mx) ? v : mx;
  }
  inter_l[wid * 64 + lane]      = sum * (1.0f / (float)NNODES);  // gmean
  inter_l[wid * 64 + 32 + lane] = mx;                            // gmax
}

// ---------------- BN over batch for lin1 output -----------------------------------
__global__ void bn1_kernel(const float* __restrict__ Y, const float* __restrict__ gamma,
                           const float* __restrict__ beta, float* __restrict__ st) {
  int c = blockIdx.x * blockDim.x + threadIdx.x;
  if (c >= N1) return;
  float s = 0.0f, q = 0.0f;
  for (int b = 0; b < NB; ++b) { float v = Y[b * N1 + c]; s += v; q += v * v; }
  float mu  = s / (float)NB;
  float var = q / (float)NB - mu * mu;
  float sc  = rsqrtf(var + 1e-5f) * gamma[c];
  st[c]      = sc;
  st[N1 + c] = beta[c] - mu * sc;
}

// ---------------- concat + relu + lin2 --------------------------------------------
__global__ void final_kernel(const float* __restrict__ Y, const float* __restrict__ st,
                             const float* __restrict__ inter, const float* __restrict__ W2,
                             const float* __restrict__ b2, float* __restrict__ out) {
  __shared__ float cat[TMPF];
  int b = blockIdx.x;
  for (int t = threadIdx.x; t < TMPF; t += blockDim.x) {
    float v;
    if (t < N1) {
      v = Y[b * N1 + t] * st[t] + st[N1 + t];
    } else {
      int j = t - N1, l = j >> 6, k = j & 63;
      v = inter[l * NB * 64 + b * 64 + k];     // gmean already divided
    }
    cat[t] = (v > 0.0f) ? v : 0.0f;
  }
  __syncthreads();
  for (int o = threadIdx.x; o < NCLS; o += blockDim.x) {
    float acc = b2[o];
    for (int j = 0; j < TMPF; ++j) acc += cat[j] * W2[j * NCLS + o];
    out[b * NCLS + o] = acc;
  }
}

// ---------------- host launcher ----------------------------------------------------
extern "C" void kernel_launch(void* const* d_in, const int* in_sizes, int n_in,
                              void* d_out, int out_size, void* d_ws, size_t ws_size,
                              hipStream_t stream) {
  const float* x        = (const float*)d_in[0];
  const int*   ei       = (const int*)  d_in[1];
  const float* conv_W   = (const float*)d_in[3];
  const float* att_l    = (const float*)d_in[4];
  const float* att_r    = (const float*)d_in[5];
  const float* conv_b   = (const float*)d_in[6];
  const float* bn_gamma = (const float*)d_in[7];
  const float* bn_beta  = (const float*)d_in[8];
  const float* lin1_W   = (const float*)d_in[9];
  const float* lin1_b   = (const float*)d_in[10];
  const float* bn1_g    = (const float*)d_in[11];
  const float* bn1_b    = (const float*)d_in[12];
  const float* lin2_W   = (const float*)d_in[13];
  const float* lin2_b   = (const float*)d_in[14];
  float* out = (float*)d_out;

  const int* srcA = ei;
  const int* dstA = ei + EDG;

  // workspace layout (floats)
  float* ws    = (float*)d_ws;
  float* H     = ws;                       // NTOT*128
  float* AGG   = H     + NTOT * HC;        // NTOT*128
  float* XBUF  = AGG   + NTOT * HC;        // NTOT*32
  float* ALPHA = XBUF  + NTOT * NC;        // ETOT*4
  float* AMAX  = ALPHA + ETOT * NH;        // NTOT*4
  float* DEN   = AMAX  + NTOT * NH;        // NTOT*4
  float* STATS = DEN   + NTOT * NH;        // 64
  float* PART  = STATS + 64;               // 64*64
  float* INTER = PART  + 64 * 64;          // 3*256*64
  float* Y1    = INTER + NL * NB * 64;     // 256*124
  float* ST1   = Y1    + NB * N1;          // 248
  float* Y1P   = ST1   + 2 * N1;           // KSPLIT*256*128

  auto fill = [&](float* p, int n, float v) {
    fill_kernel<<<(n + 255) / 256, 256, 0, stream>>>(p, n, v);
  };

  const float* xin = x;
  for (int l = 0; l < NL; ++l) {
    fill(AMAX, NTOT * NH, -INFINITY);
    fill(DEN, NTOT * NH, 0.0f);
    fill(AGG, NTOT * HC, 0.0f);

    gemm_xw_kernel<<<NTOT / 16, 256, 0, stream>>>(xin, conv_W + l * ND * HC, H, NTOT / 16);

    edge_alpha_kernel<<<ETOT / 8, 256, 0, stream>>>(H, srcA, dstA,
        att_l + l * NH * NC, att_r + l * NH * NC, ALPHA, AMAX);

    edge_exp_kernel<<<(ETOT * NH + 255) / 256, 256, 0, stream>>>(dstA, ALPHA, AMAX, DEN);

    edge_agg_kernel<<<ETOT / 8, 256, 0, stream>>>(H, srcA, dstA, ALPHA, DEN, AGG);

    node_post_kernel<<<64, 256, 0, stream>>>(AGG, conv_b + l * NC, XBUF, PART);

    bn_finalize_kernel<<<1, 32, 0, stream>>>(PART, bn_gamma + l * NC, bn_beta + l * NC, STATS);

    bn_relu_pool_kernel<<<NB / 8, 256, 0, stream>>>(XBUF, STATS, INTER + l * NB * 64);

    xin = XBUF;
  }

  // lin1: view XBUF as [256, 3968], K split 16 ways for occupancy
  gemm_lin1_kernel<<<(16 * 8 * KSPLIT) / 8, 256, 0, stream>>>(XBUF, lin1_W, Y1P);
  lin1_reduce_kernel<<<(NB * N1 + 255) / 256, 256, 0, stream>>>(Y1P, lin1_b, Y1);
  bn1_kernel<<<1, 128, 0, stream>>>(Y1, bn1_g, bn1_b, ST1);
  final_kernel<<<NB, 128, 0, stream>>>(Y1, ST1, INTER, lin2_W, lin2_b, out);
}